// Net_3341484556670
// MI455X (gfx1250) — compile-verified
//
#include <hip/hip_runtime.h>
#include <hip/hip_bf16.h>

typedef __attribute__((ext_vector_type(2))) float v2f;
typedef __attribute__((ext_vector_type(8))) float v8f;

#define ZF 46  // final embedding width (23 + 23)

static inline int cdiv(int a, int b) { return (a + b - 1) / b; }

// ---------------------------------------------------------------- utilities

__global__ void fillf_kernel(float* __restrict__ p, float v, long long n) {
  long long i = (long long)blockIdx.x * blockDim.x + threadIdx.x;
  if (i < n) p[i] = v;
}

__global__ void filli_kernel(int* __restrict__ p, int v, int n) {
  int i = blockIdx.x * blockDim.x + threadIdx.x;
  if (i < n) p[i] = v;
}

// dst = src, float4 granularity (all sizes here are multiples of 4 floats)
__global__ void copyf4_kernel(float4* __restrict__ dst, const float4* __restrict__ src,
                              int n4) {
  int i = blockIdx.x * blockDim.x + threadIdx.x;
  if (i < n4) dst[i] = src[i];
}

// deg[dst] += 1 per edge
__global__ void count_deg_kernel(const int* __restrict__ dst, int E, int* __restrict__ deg) {
  int i = blockIdx.x * blockDim.x + threadIdx.x;
  if (i < E) atomicAdd(&deg[dst[i]], 1);
}

// rdeg = 1/(indeg+1)   (self-loop included in mean)
__global__ void make_rdeg_kernel(const int* __restrict__ deg, float* __restrict__ rdeg, int N) {
  int i = blockIdx.x * blockDim.x + threadIdx.x;
  if (i < N) rdeg[i] = 1.0f / (float)(deg[i] + 1);
}

// O[n, 0:F] = A[n, :], O[n, F:2F] = B[n, :]
__global__ void concat2_kernel(const float* __restrict__ A, const float* __restrict__ B,
                               float* __restrict__ O, int N, int F) {
  int idx = blockIdx.x * blockDim.x + threadIdx.x;
  int total = N * 2 * F;
  if (idx >= total) return;
  int n = idx / (2 * F);
  int f = idx - n * 2 * F;
  O[idx] = (f < F) ? A[(size_t)n * F + f] : B[(size_t)n * F + (f - F)];
}

// acc[dst[e], 2f:2f+2] += X[src[e], 2f:2f+2]  (F is even for every layer)
// float2 gather read + two hardware f32 atomics into the L2-resident accumulator.
__global__ void scatter_add2_kernel(const float* __restrict__ X, float* __restrict__ acc,
                                    const int* __restrict__ src, const int* __restrict__ dst,
                                    int E, int F) {
  int F2 = F >> 1;
  int idx = blockIdx.x * blockDim.x + threadIdx.x;
  int total = E * F2;
  if (idx >= total) return;
  int e = idx / F2;
  int f2 = idx - e * F2;
  int s = src[e], d = dst[e];
  const float2 v = ((const float2*)(X + (size_t)s * F))[f2];
  float* a = acc + (size_t)d * F + 2 * f2;
  unsafeAtomicAdd(a + 0, v.x);
  unsafeAtomicAdd(a + 1, v.y);
}

// ---------------------------------------------------------------- WMMA GEMM
// ACC already holds msg_sum = x + sum_neighbors(x).
// H = sigmoid( (ACC * rdeg[row]) @ W + b ),  W: [Fin, Fout] row-major.
// One wave32 computes a 16-row strip x CT 16-col tiles via V_WMMA_F32_16X16X4_F32:
// A (16x4 f32, 2 VGPRs) is loaded once per k-step and feeds CT WMMAs.
template <int CT>
__global__ void __launch_bounds__(256)
sage_gemm_sigmoid_kernel(const float* __restrict__ ACC, const float* __restrict__ rdeg,
                         const float* __restrict__ W, const float* __restrict__ bias,
                         float* __restrict__ H, int N, int Fin, int Fout) {
  int gtid = blockIdx.x * blockDim.x + threadIdx.x;
  int wave = gtid >> 5;                 // wave-uniform
  int lane = threadIdx.x & 31;
  int rowTiles = (N + 15) >> 4;
  if (wave >= rowTiles) return;         // whole-wave exit: EXEC all-1 for WMMA

  int row0 = wave << 4;
  int half = lane >> 4;                 // A: lanes 0-15 hold K=k0..k0+1, lanes 16-31 K=k0+2..k0+3
  int l16  = lane & 15;

  int arow = row0 + l16;
  float ascale = (arow < N) ? rdeg[arow] : 0.0f;
  int arow_c = (arow < N) ? arow : (N - 1);
  const float* accr = ACC + (size_t)arow_c * Fin;

  // B column pointers (clamped; out-of-range cols compute garbage, never stored)
  const float* wt[CT];
#pragma unroll
  for (int t = 0; t < CT; ++t) {
    int bc = t * 16 + l16;
    wt[t] = W + ((bc < Fout) ? bc : (Fout - 1));
  }

  v8f c[CT] = {};
  int Fin4 = Fin & ~3;
#pragma unroll 2
  for (int k0 = 0; k0 < Fin4; k0 += 4) {
    int ka = k0 + half * 2;
    const float2 av = *(const float2*)(accr + ka);  // 8B-aligned: Fin even, ka even
    v2f a;
    a.x = av.x * ascale;
    a.y = av.y * ascale;
#pragma unroll
    for (int t = 0; t < CT; ++t) {
      v2f b;
      b.x = wt[t][(size_t)ka * Fout];
      b.y = wt[t][(size_t)(ka + 1) * Fout];
      c[t] = __builtin_amdgcn_wmma_f32_16x16x4_f32(false, a, false, b, (short)0, c[t],
                                                   false, false);
    }
  }
  if (Fin4 < Fin) {  // ragged tail (Fin=34): masked, branch-free per lane
    int ka = Fin4 + half * 2;
    int k0c = (ka < Fin) ? ka : (Fin - 1);
    int k1c = (ka + 1 < Fin) ? (ka + 1) : (Fin - 1);
    float m0 = (ka < Fin) ? ascale : 0.0f;
    float m1 = (ka + 1 < Fin) ? ascale : 0.0f;
    v2f a;
    a.x = accr[k0c] * m0;
    a.y = accr[k1c] * m1;
#pragma unroll
    for (int t = 0; t < CT; ++t) {
      v2f b;
      b.x = wt[t][(size_t)k0c * Fout];
      b.y = wt[t][(size_t)k1c * Fout];
      c[t] = __builtin_amdgcn_wmma_f32_16x16x4_f32(false, a, false, b, (short)0, c[t],
                                                   false, false);
    }
  }

  // D layout: VGPR j holds (M = j + 8*half, Ncol = lane%16)
#pragma unroll
  for (int t = 0; t < CT; ++t) {
    int col = t * 16 + l16;
    if (col < Fout) {
      float bv = bias[col];
      for (int j = 0; j < 8; ++j) {
        int r = row0 + j + 8 * half;
        if (r < N) {
          float v = c[t][j] + bv;
          H[(size_t)r * Fout + col] = 1.0f / (1.0f + __expf(-v));
        }
      }
    }
  }
}

// ---------------------------------------------------------------- reductions

__device__ __forceinline__ float blockReduceSum(float v) {
  __shared__ float sm[32];
  for (int o = 16; o > 0; o >>= 1) v += __shfl_down(v, o, 32);
  int lane = threadIdx.x & 31, wid = threadIdx.x >> 5;
  if (lane == 0) sm[wid] = v;
  __syncthreads();
  int nw = (blockDim.x + 31) >> 5;
  v = (threadIdx.x < nw) ? sm[threadIdx.x] : 0.0f;
  if (wid == 0)
    for (int o = 16; o > 0; o >>= 1) v += __shfl_down(v, o, 32);
  return v;
}

// sum over edges of max(||z_i - z_j||^2 - ||z_i - z_k||^2, 0)
__global__ void triplet_kernel(const float* __restrict__ z, const int* __restrict__ ii,
                               const int* __restrict__ jj, const int* __restrict__ kk,
                               int E, float* __restrict__ acc) {
  int e = blockIdx.x * blockDim.x + threadIdx.x;
  float v = 0.0f;
  if (e < E) {
    const float* zi = z + (size_t)ii[e] * ZF;
    const float* zj = z + (size_t)jj[e] * ZF;
    const float* zk = z + (size_t)kk[e] * ZF;
    float s1 = 0.0f, s2 = 0.0f;
    for (int f = 0; f < ZF; ++f) {
      float zif = zi[f];
      float dij = zif - zj[f];
      float dik = zif - zk[f];
      s1 += dij * dij;
      s2 += dik * dik;
    }
    v = fmaxf(s1 - s2, 0.0f);
  }
  v = blockReduceSum(v);
  if (threadIdx.x == 0) unsafeAtomicAdd(acc, v);
}

// sum over 3Ep+3En rows of -log_softmax(cat(zl, zr) @ reg_w)[target]
__global__ void regloss_kernel(const float* __restrict__ z, const float* __restrict__ rw,
                               const int* __restrict__ pe, const int* __restrict__ ne,
                               const int* __restrict__ psur, const int* __restrict__ nsur,
                               const int* __restrict__ target, int Ep, int En,
                               float* __restrict__ acc) {
  __shared__ float srw[2 * ZF * 3];  // 276 floats, broadcast from LDS
  for (int i = threadIdx.x; i < 2 * ZF * 3; i += blockDim.x) srw[i] = rw[i];
  __syncthreads();

  int r = blockIdx.x * blockDim.x + threadIdx.x;
  int total = 3 * Ep + 3 * En;
  float v = 0.0f;
  if (r < total) {
    int li, ri, r2 = r;
    // block order: (pzi,pzj) (nzi,nzj) (nzi,nzk) (nzj,nzk) (pzi,pzk) (pzj,pzk)
    if (r2 < Ep) { li = pe[r2]; ri = pe[Ep + r2]; }
    else { r2 -= Ep;
      if (r2 < En) { li = ne[r2]; ri = ne[En + r2]; }
      else { r2 -= En;
        if (r2 < En) { li = ne[r2]; ri = nsur[r2]; }
        else { r2 -= En;
          if (r2 < En) { li = ne[En + r2]; ri = nsur[r2]; }
          else { r2 -= En;
            if (r2 < Ep) { li = pe[r2]; ri = psur[r2]; }
            else { r2 -= Ep; li = pe[Ep + r2]; ri = psur[r2]; }
          } } } }
    const float* L = z + (size_t)li * ZF;
    const float* R = z + (size_t)ri * ZF;
    float l0 = 0.0f, l1 = 0.0f, l2 = 0.0f;
    for (int f = 0; f < ZF; ++f) {
      float x = L[f];
      l0 += x * srw[f * 3 + 0]; l1 += x * srw[f * 3 + 1]; l2 += x * srw[f * 3 + 2];
    }
    for (int f = 0; f < ZF; ++f) {
      float x = R[f];
      l0 += x * srw[(ZF + f) * 3 + 0]; l1 += x * srw[(ZF + f) * 3 + 1];
      l2 += x * srw[(ZF + f) * 3 + 2];
    }
    float m = fmaxf(l0, fmaxf(l1, l2));
    float s = __expf(l0 - m) + __expf(l1 - m) + __expf(l2 - m);
    int t = target[r];
    float lt = (t == 0) ? l0 : ((t == 1) ? l1 : l2);
    v = -((lt - m) - logf(s));
  }
  v = blockReduceSum(v);
  if (threadIdx.x == 0) unsafeAtomicAdd(acc, v);
}

__global__ void finalize_kernel(const float* __restrict__ scal, float* __restrict__ out,
                                int Ep, int En) {
  if (blockIdx.x == 0 && threadIdx.x == 0) {
    float l1  = scal[0] / (float)Ep;
    float l2  = scal[1] / (float)En;
    float reg = scal[2] / (float)(3 * Ep + 3 * En);
    out[0] = reg + 0.1f * (l1 + l2);
  }
}

// ---------------------------------------------------------------- driver

extern "C" void kernel_launch(void* const* d_in, const int* in_sizes, int n_in,
                              void* d_out, int out_size, void* d_ws, size_t ws_size,
                              hipStream_t stream) {
  const float* d1   = (const float*)d_in[0];
  const float* d2   = (const float*)d_in[1];
  const int*   pe   = (const int*)d_in[2];   // [2,Ep]: src=pe, dst=pe+Ep
  const int*   ne   = (const int*)d_in[3];
  const int*   psur = (const int*)d_in[4];
  const int*   nsur = (const int*)d_in[5];
  const int*   tgt  = (const int*)d_in[6];
  const float* Wp[4] = {(const float*)d_in[7],  (const float*)d_in[11], (const float*)d_in[15], (const float*)d_in[19]};
  const float* Bp[4] = {(const float*)d_in[8],  (const float*)d_in[12], (const float*)d_in[16], (const float*)d_in[20]};
  const float* Wn[4] = {(const float*)d_in[9],  (const float*)d_in[13], (const float*)d_in[17], (const float*)d_in[21]};
  const float* Bn[4] = {(const float*)d_in[10], (const float*)d_in[14], (const float*)d_in[18], (const float*)d_in[22]};
  const float* regw  = (const float*)d_in[23];

  const int N  = in_sizes[0] / 128;
  const int Ep = in_sizes[2] / 2;
  const int En = in_sizes[3] / 2;
  const int FIN[4]  = {128, 76, 34, 80};
  const int FOUT[4] = {38, 17, 40, 23};

  // workspace layout (floats, 16B-aligned slots)
  float* w = (float*)d_ws;
  size_t off = 0;
  auto alloc = [&](size_t nf) { float* p = w + off; off += (nf + 3) & ~(size_t)3; return p; };
  float* rdegp = alloc(N);
  float* rdegn = alloc(N);
  int*   deg   = (int*)alloc(N);
  float* hA    = alloc((size_t)N * 40);   // hp
  float* hB    = alloc((size_t)N * 40);   // hn
  float* xp    = alloc((size_t)N * 80);   // concat(hp,hn)
  float* xn    = alloc((size_t)N * 80);   // concat(hn,hp)
  float* accp  = alloc((size_t)N * 128);
  float* accn  = alloc((size_t)N * 128);
  float* zb    = alloc((size_t)N * ZF);
  float* scal  = alloc(8);                // [loss1, loss2, regsum]

  const int T = 256;

  // degrees (once per edge set, reused across all 4 layers)
  filli_kernel<<<cdiv(N, T), T, 0, stream>>>(deg, 0, N);
  count_deg_kernel<<<cdiv(Ep, T), T, 0, stream>>>(pe + Ep, Ep, deg);
  make_rdeg_kernel<<<cdiv(N, T), T, 0, stream>>>(deg, rdegp, N);
  filli_kernel<<<cdiv(N, T), T, 0, stream>>>(deg, 0, N);
  count_deg_kernel<<<cdiv(En, T), T, 0, stream>>>(ne + En, En, deg);
  make_rdeg_kernel<<<cdiv(N, T), T, 0, stream>>>(deg, rdegn, N);

  fillf_kernel<<<1, 32, 0, stream>>>(scal, 0.0f, 8);

  const float* inP = d1;
  const float* inN = d2;
  for (int i = 0; i < 4; ++i) {
    int fin = FIN[i], fout = FOUT[i];
    if (i > 0) {
      int tot = N * fin;
      concat2_kernel<<<cdiv(tot, T), T, 0, stream>>>(hA, hB, xp, N, fin / 2);
      concat2_kernel<<<cdiv(tot, T), T, 0, stream>>>(hB, hA, xn, N, fin / 2);
      inP = xp; inN = xn;
    }
    int gemmBlocks = cdiv(cdiv(N, 16) * 32, T);
    int colTiles = cdiv(fout, 16);
    // positive branch: acc = x (self-loop), then += neighbor features
    copyf4_kernel<<<cdiv(N * fin / 4, T), T, 0, stream>>>((float4*)accp, (const float4*)inP,
                                                          N * fin / 4);
    scatter_add2_kernel<<<cdiv(Ep * fin / 2, T), T, 0, stream>>>(inP, accp, pe, pe + Ep, Ep, fin);
    if (colTiles == 2)
      sage_gemm_sigmoid_kernel<2><<<gemmBlocks, T, 0, stream>>>(accp, rdegp, Wp[i], Bp[i], hA, N, fin, fout);
    else
      sage_gemm_sigmoid_kernel<3><<<gemmBlocks, T, 0, stream>>>(accp, rdegp, Wp[i], Bp[i], hA, N, fin, fout);
    // negative branch
    copyf4_kernel<<<cdiv(N * fin / 4, T), T, 0, stream>>>((float4*)accn, (const float4*)inN,
                                                          N * fin / 4);
    scatter_add2_kernel<<<cdiv(En * fin / 2, T), T, 0, stream>>>(inN, accn, ne, ne + En, En, fin);
    if (colTiles == 2)
      sage_gemm_sigmoid_kernel<2><<<gemmBlocks, T, 0, stream>>>(accn, rdegn, Wn[i], Bn[i], hB, N, fin, fout);
    else
      sage_gemm_sigmoid_kernel<3><<<gemmBlocks, T, 0, stream>>>(accn, rdegn, Wn[i], Bn[i], hB, N, fin, fout);
  }

  // z = concat(hp, hn) : [N, 46]
  concat2_kernel<<<cdiv(N * ZF, T), T, 0, stream>>>(hA, hB, zb, N, ZF / 2);

  // loss1 = mean max(||zi-zj||^2 - ||zi-zk||^2, 0) over positive edges
  triplet_kernel<<<cdiv(Ep, T), T, 0, stream>>>(zb, pe, pe + Ep, psur, Ep, scal + 0);
  // loss2 = mean max(||zi-zk||^2 - ||zi-zj||^2, 0) over negative edges (roles swapped)
  triplet_kernel<<<cdiv(En, T), T, 0, stream>>>(zb, ne, nsur, ne + En, En, scal + 1);

  int totalRows = 3 * Ep + 3 * En;
  regloss_kernel<<<cdiv(totalRows, T), T, 0, stream>>>(zb, regw, pe, ne, psur, nsur, tgt,
                                                       Ep, En, scal + 2);

  finalize_kernel<<<1, 32, 0, stream>>>(scal, (float*)d_out, Ep, En);
}